// BestDetectorEverLoss_25374666785519
// MI455X (gfx1250) — compile-verified
//
#include <hip/hip_runtime.h>
#include <hip/hip_bf16.h>
#include <math.h>

typedef __attribute__((ext_vector_type(2))) float v2f;
typedef __attribute__((ext_vector_type(8))) float v8f;

#define GG 7
#define CELLS 49
#define ANC 3
#define WPB 8          // waves per block (256 threads)
#define MAXBLK 1024    // max partials written to d_ws

__device__ __forceinline__ float bce_f(float p, float t) {
    // -(t*log(p) + (1-t)*log(1-p)), matching reference formulation
    return -(t * logf(p) + (1.0f - t) * logf(1.0f - p));
}

__global__ __launch_bounds__(256)
void detloss_main(const float* __restrict__ bboxp,   // (N,15,7,7) predictions
                  const float* __restrict__ clsp,    // (N,2) class logits
                  const float* __restrict__ bboxg,   // (N,5,7,7) ground truth
                  const int*   __restrict__ clsg,    // (N,) labels in {1,2}
                  float* __restrict__ blocksums,
                  int Ntot, int nGroups)
{
    __shared__ float sm[WPB][ANC * CELLS];   // async-prefetched pred prob planes
    __shared__ float swsum[WPB];
    const int lane = threadIdx.x & 31;
    const int wv   = threadIdx.x >> 5;

    float part = 0.0f;  // per-lane running partial across all samples this wave handles

    for (int g = blockIdx.x; g < nGroups; g += gridDim.x) {
        const int n = g * WPB + wv;        // wave-uniform
        if (n >= Ntot) continue;

        const float* Bp = bboxp + (size_t)n * 735;  // 15*49
        const float* Gt = bboxg + (size_t)n * 245;  // 5*49
        float* myl = sm[wv];

        // ---- async prefetch: pred prob planes (channels 0,5,10) -> LDS ----
        #pragma unroll
        for (int a = 0; a < ANC; ++a) {
            {
                unsigned lds0 = (unsigned)(unsigned long long)(void*)&myl[a * CELLS + lane];
                unsigned long long g0 = (unsigned long long)(Bp + a * 245 + lane);
                asm volatile("global_load_async_to_lds_b32 %0, %1, off"
                             :: "v"(lds0), "v"(g0) : "memory");
            }
            if (lane < CELLS - 32) {
                unsigned lds1 = (unsigned)(unsigned long long)(void*)&myl[a * CELLS + 32 + lane];
                unsigned long long g1 = (unsigned long long)(Bp + a * 245 + 32 + lane);
                asm volatile("global_load_async_to_lds_b32 %0, %1, off"
                             :: "v"(lds1), "v"(g1) : "memory");
            }
        }

        // ---- GT prob plane + argmax cell (first-index tie break) ----
        float p0 = Gt[lane];
        float p1 = -1.0f;
        if (lane < CELLS - 32) p1 = Gt[32 + lane];

        float bv = p0; int bi = lane;
        if (lane < CELLS - 32 && p1 > bv) { bv = p1; bi = 32 + lane; }
        #pragma unroll
        for (int off = 16; off >= 1; off >>= 1) {
            float ov = __shfl_xor(bv, off, 32);
            int   oi = __shfl_xor(bi, off, 32);
            if (ov > bv || (ov == bv && oi < bi)) { bv = ov; bi = oi; }
        }
        const int m  = bi;
        const int ii = m / GG, jj = m % GG;

        // ---- scattered gathers distributed over lanes, then broadcast ----
        // lanes 0-3 : GT coords ch 1..4 ; lanes 4-15 : pred coords ch {1..4,6..9,11..14}
        float tmp = 0.0f;
        if (lane < 4) {
            tmp = Gt[(lane + 1) * CELLS + m];
        } else if (lane < 16) {
            int t  = lane - 4;
            int ch = 1 + (t >> 2) * 5 + (t & 3);
            tmp = Bp[ch * CELLS + m];
        }
        float gx = __shfl(tmp, 0, 32);
        float gy = __shfl(tmp, 1, 32);
        float gw = __shfl(tmp, 2, 32);
        float gh = __shfl(tmp, 3, 32);
        float rx[ANC], ry[ANC], rw[ANC], rh[ANC];
        #pragma unroll
        for (int a = 0; a < ANC; ++a) {
            rx[a] = __shfl(tmp, 4 + 4 * a + 0, 32);
            ry[a] = __shfl(tmp, 4 + 4 * a + 1, 32);
            rw[a] = __shfl(tmp, 4 + 4 * a + 2, 32);
            rh[a] = __shfl(tmp, 4 + 4 * a + 3, 32);
        }

        // ---- IoU anchor selection (redundant scalar work on all lanes) ----
        const float jf = (float)jj, if_ = (float)ii;
        const float tx = (gx + jf) / 7.0f;
        const float ty = (gy + if_) / 7.0f;
        const float tx1 = tx - gw * 0.5f, ty1 = ty - gh * 0.5f;
        const float tx2 = tx + gw * 0.5f, ty2 = ty + gh * 0.5f;
        const float tarea = (tx2 - tx1) * (ty2 - ty1);
        int best = 0; float bestiou = -1.0f;
        #pragma unroll
        for (int a = 0; a < ANC; ++a) {
            float ax  = (rx[a] + jf) / 7.0f;
            float ay  = (ry[a] + if_) / 7.0f;
            float ax1 = ax - rw[a] * 0.5f, ay1 = ay - rh[a] * 0.5f;
            float ax2 = ax + rw[a] * 0.5f, ay2 = ay + rh[a] * 0.5f;
            float iw = fmaxf(fminf(ax2, tx2) - fmaxf(ax1, tx1), 0.0f);
            float ih = fmaxf(fminf(ay2, ty2) - fmaxf(ay1, ty1), 0.0f);
            float inter = iw * ih;
            float uni   = (ax2 - ax1) * (ay2 - ay1) + tarea - inter;
            float iou   = inter / (uni + 1e-9f);
            if (iou > bestiou) { bestiou = iou; best = a; }
        }

        // ---- prob loss: consume async-prefetched planes from LDS ----
        asm volatile("s_wait_asynccnt 0x0" ::: "memory");
        float acc = 0.0f;
        #pragma unroll
        for (int a = 0; a < ANC; ++a) {
            float t0 = (a == best) ? p0 : 0.0f;
            acc += bce_f(myl[a * CELLS + lane], t0);
            if (lane < CELLS - 32) {
                float t1 = (a == best) ? p1 : 0.0f;
                acc += bce_f(myl[a * CELLS + 32 + lane], t1);
            }
        }
        part += acc * (1.0f / ((float)Ntot * (float)CELLS));

        if (lane == 0) {
            // coord + size for best anchor (raw cell values, per reference)
            float coord = bce_f(rx[best], gx) + bce_f(ry[best], gy);
            float size  = fabsf(logf(rw[best]) - logf(gw))
                        + fabsf(logf(rh[best]) - logf(gh));
            // cross entropy: mean(lse - selected_logit)
            float x0 = clsp[2 * (size_t)n + 0];
            float x1 = clsp[2 * (size_t)n + 1];
            int   k  = clsg[n] - 1;
            float mx  = fmaxf(x0, x1);
            float lse = mx + logf(expf(x0 - mx) + expf(x1 - mx));
            float sel = (k == 0) ? x0 : x1;
            part += coord + size + (lse - sel) * (1.0f / (float)Ntot);
        }
    }

    // ---- deterministic wave + block reduction (once per block) ----
    #pragma unroll
    for (int off = 16; off >= 1; off >>= 1) part += __shfl_xor(part, off, 32);
    if (lane == 0) swsum[wv] = part;
    __syncthreads();
    if (threadIdx.x == 0) {
        float s = 0.0f;
        #pragma unroll
        for (int w = 0; w < WPB; ++w) s += swsum[w];
        blocksums[blockIdx.x] = s;
    }
}

// Final reduction: one wave, pipelined loads (4 independent accumulators),
// then lane partials collapsed with a v_wmma_f32_16x16x4_f32 ones-product:
// A[m][0]=acc(lane m), A[m][2]=acc(lane m+16), B = all ones
//   =>  every D column holds row-sums of A.
__global__ __launch_bounds__(32)
void detloss_reduce(const float* __restrict__ bs, float* __restrict__ out, int P)
{
    const int lane = threadIdx.x;
    float a0 = 0.0f, a1 = 0.0f, a2 = 0.0f, a3 = 0.0f;
    for (int i = lane; i < P; i += 128) {
        a0 += bs[i];
        if (i + 32 < P) a1 += bs[i + 32];
        if (i + 64 < P) a2 += bs[i + 64];
        if (i + 96 < P) a3 += bs[i + 96];
    }
    float acc = (a0 + a1) + (a2 + a3);

    v2f a; a[0] = acc;  a[1] = 0.0f;
    v2f b; b[0] = 1.0f; b[1] = 1.0f;
    v8f c = {};
    c = __builtin_amdgcn_wmma_f32_16x16x4_f32(
            /*neg_a=*/false, a, /*neg_b=*/false, b,
            /*c_mod=*/(short)0, c, /*reuse_a=*/false, /*reuse_b=*/false);
    // lanes 0-15 hold rowsums of M=0..7 in c[0..7]; lanes 16-31 hold M=8..15
    float r = c[0] + c[1] + c[2] + c[3] + c[4] + c[5] + c[6] + c[7];
    r += __shfl_xor(r, 16, 32);
    if (lane == 0) out[0] = r;
}

extern "C" void kernel_launch(void* const* d_in, const int* in_sizes, int n_in,
                              void* d_out, int out_size, void* d_ws, size_t ws_size,
                              hipStream_t stream)
{
    const float* bboxp = (const float*)d_in[0];  // bbox_ (N,15,7,7)
    const float* clsp  = (const float*)d_in[1];  // cls_  (N,2)
    const float* bboxg = (const float*)d_in[2];  // bbox  (N,5,7,7)
    const int*   clsg  = (const int*)d_in[3];    // cls   (N,)
    const int N      = in_sizes[3];
    const int groups = (N + WPB - 1) / WPB;
    const int nblk   = groups < MAXBLK ? groups : MAXBLK;

    float* bs = (float*)d_ws;                    // nblk floats of scratch
    detloss_main<<<nblk, 256, 0, stream>>>(bboxp, clsp, bboxg, clsg, bs, N, groups);
    detloss_reduce<<<1, 32, 0, stream>>>(bs, (float*)d_out, nblk);
}